// Fredkin3plus6Layer_68590627717483
// MI455X (gfx1250) — compile-verified
//
#include <hip/hip_runtime.h>
#include <math.h>

#define G_GROUPS 1024
#define DIN_     3072          // x columns; also output row length (G*3)
#define THREADS_ 1024          // one thread per group
#define CHUNKS_  768           // 3072 floats / 4 per 16B chunk
#define RPB      8             // rows per block (amortize coef loads)

typedef int v4i __attribute__((ext_vector_type(4)));
typedef __attribute__((address_space(1))) v4i as1_v4i;
typedef __attribute__((address_space(3))) v4i as3_v4i;

// ---------------- CDNA5 async global->LDS helpers -------------------------

__device__ __forceinline__ void issue_async_b128(const float* gsrc, float* ldst) {
#if __has_builtin(__builtin_amdgcn_global_load_async_to_lds_b128)
  __builtin_amdgcn_global_load_async_to_lds_b128(
      (as1_v4i*)gsrc, (as3_v4i*)ldst, /*offset=*/0, /*cpol=*/0);
#else
  unsigned laddr = (unsigned)(size_t)(__attribute__((address_space(3))) float*)ldst;
  asm volatile("global_load_async_to_lds_b128 %0, %1, off"
               :: "v"(laddr), "v"(gsrc) : "memory");
#endif
}

__device__ __forceinline__ void wait_async0() {
#if __has_builtin(__builtin_amdgcn_s_wait_asynccnt)
  __builtin_amdgcn_s_wait_asynccnt(0);
#else
  asm volatile("s_wait_asynccnt 0" ::: "memory");
#endif
}

__device__ __forceinline__ void wait_async1() {
#if __has_builtin(__builtin_amdgcn_s_wait_asynccnt)
  __builtin_amdgcn_s_wait_asynccnt(1);
#else
  asm volatile("s_wait_asynccnt 1" ::: "memory");
#endif
}

// ---------------- per-group coefficient derivation ------------------------
// out[b,g,m] = K[m] + sum_t L[m][t]*x_t + sum_q Q[m][q]*prod_q,
// prods: q=0 -> x0*x1, q=1 -> x0*x2, q=2 -> x1*x2.

__device__ __forceinline__ void group_coefs(
    const int* __restrict__ conn, const float* __restrict__ sel,
    const float* __restrict__ wg, int g,
    int idx[3], float K[3], float L[3][3], float Q[3][3])
{
  float a[3], bb[3];
#pragma unroll
  for (int j = 0; j < 3; ++j) {
    idx[j]  = conn[g * 3 + j];
    float s = sel[g * 3 + j];
    float C = s / (1.0f + fabsf(s));
    float ac = fabsf(C);
    a[j]  = 1.0f - ac;          // gate slope
    bb[j] = 0.5f * (C + ac);    // gate offset
  }
  // softmax over the 6 permutation weights
  float wv[6];
  float mx = -3.4e38f;
#pragma unroll
  for (int p = 0; p < 6; ++p) { wv[p] = wg[g * 6 + p]; mx = fmaxf(mx, wv[p]); }
  float sum = 0.0f;
#pragma unroll
  for (int p = 0; p < 6; ++p) { wv[p] = __expf(wv[p] - mx); sum += wv[p]; }
  float inv = 1.0f / sum;
#pragma unroll
  for (int p = 0; p < 6; ++p) wv[p] *= inv;

  // accumulate polynomial coefficients in the *gated* variables
  // perm p=(i,j,k): m0 += w*g_i ; m1 += w*(g_k + g_i g_j - g_i g_k)
  //                 m2 += w*(g_j - g_i g_j + g_i g_k)
  const int P0[6] = {0, 0, 1, 1, 2, 2};
  const int P1[6] = {1, 2, 0, 2, 0, 1};
  const int P2[6] = {2, 1, 2, 0, 1, 0};
  float lin[3][3]  = {{0}};
  float quad[3][3] = {{0}};
#pragma unroll
  for (int p = 0; p < 6; ++p) {
    int i = P0[p], j = P1[p], k = P2[p];
    float w = wv[p];
    lin[0][i] += w;
    lin[1][k] += w;
    lin[2][j] += w;
    int qij = i + j - 1;   // unordered pair {0,1}->0 {0,2}->1 {1,2}->2
    int qik = i + k - 1;
    quad[1][qij] += w; quad[1][qik] -= w;
    quad[2][qij] -= w; quad[2][qik] += w;
  }
  // fold gate affine g_t = a_t*x_t + bb_t into (K, L, Q) over raw x
  const int QA[3] = {0, 0, 1}, QB[3] = {1, 2, 2};
#pragma unroll
  for (int m = 0; m < 3; ++m) {
    float k0 = 0.0f;
#pragma unroll
    for (int t = 0; t < 3; ++t) { L[m][t] = lin[m][t] * a[t]; k0 += lin[m][t] * bb[t]; }
#pragma unroll
    for (int q = 0; q < 3; ++q) {
      int u = QA[q], v = QB[q];
      float c = quad[m][q];
      k0      += c * bb[u] * bb[v];
      L[m][u] += c * a[u] * bb[v];
      L[m][v] += c * bb[u] * a[v];
      Q[m][q]  = c * a[u] * a[v];
    }
    K[m] = k0;
  }
}

// ---------------- kernel 1: build SoA coefficient table in d_ws -----------

__global__ void fredkin_coefs_kernel(const int* __restrict__ conn,
                                     const float* __restrict__ sel,
                                     const float* __restrict__ wg,
                                     int* __restrict__ wsi,
                                     float* __restrict__ wsf)
{
  int g = blockIdx.x * blockDim.x + threadIdx.x;
  if (g >= G_GROUPS) return;
  int idx[3]; float K[3], L[3][3], Q[3][3];
  group_coefs(conn, sel, wg, g, idx, K, L, Q);
  wsi[0 * G_GROUPS + g] = idx[0];
  wsi[1 * G_GROUPS + g] = idx[1];
  wsi[2 * G_GROUPS + g] = idx[2];
  wsf[0 * G_GROUPS + g] = K[0];
  wsf[1 * G_GROUPS + g] = K[1];
  wsf[2 * G_GROUPS + g] = K[2];
#pragma unroll
  for (int m = 0; m < 3; ++m)
#pragma unroll
    for (int q = 0; q < 3; ++q) {
      wsf[(3 + m * 3 + q)  * G_GROUPS + g] = L[m][q];
      wsf[(12 + m * 3 + q) * G_GROUPS + g] = Q[m][q];
    }
}

// ---------------- kernel 2: main bandwidth-bound pass ---------------------
// One block = 1024 threads = all groups; processes RPB rows of x.
// Rows are double-buffered in LDS via async global->LDS b128 loads.

template <bool FUSED>
__global__ __launch_bounds__(THREADS_)
void fredkin_main_kernel(const float* __restrict__ x,
                         const int* __restrict__ wsi,
                         const float* __restrict__ wsf,
                         const int* __restrict__ conn,
                         const float* __restrict__ sel,
                         const float* __restrict__ wg,
                         float* __restrict__ out,
                         long long Btot)
{
  __shared__ __align__(16) float rowbuf[2][DIN_];  // 2 x 12 KB input rows
  __shared__ __align__(16) float obuf[DIN_];       // 12 KB output repack

  const int t = threadIdx.x;          // == group id
  int i0, i1, i2;
  float cf[21];
  if (FUSED) {
    int idx[3]; float K[3], L[3][3], Q[3][3];
    group_coefs(conn, sel, wg, t, idx, K, L, Q);
    i0 = idx[0]; i1 = idx[1]; i2 = idx[2];
    cf[0] = K[0]; cf[1] = K[1]; cf[2] = K[2];
#pragma unroll
    for (int m = 0; m < 3; ++m)
#pragma unroll
      for (int q = 0; q < 3; ++q) {
        cf[3 + m * 3 + q]  = L[m][q];
        cf[12 + m * 3 + q] = Q[m][q];
      }
  } else {
    i0 = wsi[t]; i1 = wsi[G_GROUPS + t]; i2 = wsi[2 * G_GROUPS + t];
#pragma unroll
    for (int c = 0; c < 21; ++c) cf[c] = wsf[c * G_GROUPS + t];
  }

  const long long b0 = (long long)blockIdx.x * RPB;
  if (b0 >= Btot) return;

  if (t < CHUNKS_)
    issue_async_b128(x + b0 * DIN_ + t * 4, &rowbuf[0][t * 4]);

#pragma unroll 1
  for (int r = 0; r < RPB; ++r) {
    const long long b = b0 + r;
    if (b >= Btot) break;                      // uniform across block
    const bool pre = (r + 1 < RPB) && (b + 1 < Btot);
    if (pre && t < CHUNKS_)
      issue_async_b128(x + (b + 1) * DIN_ + t * 4, &rowbuf[(r + 1) & 1][t * 4]);
    if (pre) wait_async1(); else wait_async0();
    __syncthreads();                           // row r fully in LDS

    const float* row = rowbuf[r & 1];
    float x0 = row[i0], x1 = row[i1], x2 = row[i2];
    float p01 = x0 * x1, p02 = x0 * x2, p12 = x1 * x2;
    float o0 = cf[0] + cf[3] * x0 + cf[4]  * x1 + cf[5]  * x2
                      + cf[12] * p01 + cf[13] * p02 + cf[14] * p12;
    float o1 = cf[1] + cf[6] * x0 + cf[7]  * x1 + cf[8]  * x2
                      + cf[15] * p01 + cf[16] * p02 + cf[17] * p12;
    float o2 = cf[2] + cf[9] * x0 + cf[10] * x1 + cf[11] * x2
                      + cf[18] * p01 + cf[19] * p02 + cf[20] * p12;

    obuf[3 * t + 0] = o0;                      // conflict-free: gcd(3,64)=1
    obuf[3 * t + 1] = o1;
    obuf[3 * t + 2] = o2;
    __syncthreads();                           // obuf complete
    if (t < CHUNKS_) {
      float4 v = *(const float4*)(&obuf[t * 4]);
      *(float4*)(out + b * DIN_ + t * 4) = v;  // coalesced b128 stores
    }
    __syncthreads();                           // obuf reusable next iter
  }
}

// ---------------- host launch ---------------------------------------------

extern "C" void kernel_launch(void* const* d_in, const int* in_sizes, int n_in,
                              void* d_out, int out_size, void* d_ws, size_t ws_size,
                              hipStream_t stream)
{
  const float* x    = (const float*)d_in[0];
  const int*   conn = (const int*)d_in[1];
  const float* sel  = (const float*)d_in[2];
  const float* wg   = (const float*)d_in[3];
  float*       out  = (float*)d_out;

  const long long B = (long long)in_sizes[0] / DIN_;
  const int blocks  = (int)((B + RPB - 1) / RPB);
  const size_t need = (size_t)G_GROUPS * (3 * sizeof(int) + 21 * sizeof(float));

  if (d_ws && ws_size >= need) {
    int*   wsi = (int*)d_ws;
    float* wsf = (float*)((char*)d_ws + (size_t)3 * G_GROUPS * sizeof(int));
    fredkin_coefs_kernel<<<(G_GROUPS + 255) / 256, 256, 0, stream>>>(conn, sel, wg, wsi, wsf);
    fredkin_main_kernel<false><<<blocks, THREADS_, 0, stream>>>(
        x, wsi, wsf, nullptr, nullptr, nullptr, out, B);
  } else {
    fredkin_main_kernel<true><<<blocks, THREADS_, 0, stream>>>(
        x, nullptr, nullptr, conn, sel, wg, out, B);
  }
}